// FixedGraphAttentionLayer_11304353923834
// MI455X (gfx1250) — compile-verified
//
#include <hip/hip_runtime.h>

typedef float v2f __attribute__((ext_vector_type(2)));
typedef float v8f __attribute__((ext_vector_type(8)));

#define L_NODES 20000
#define BATCH   4
#define NNODES  (BATCH * L_NODES)   /* 80000 */
#define FDIM    128
#define DEG     16
#define LRELU_ALPHA 0.2f

// ---------------------------------------------------------------------------
// Kernel 1: y[NNODES x 128] = x[NNODES x 128] @ W[128 x 128], fp32 WMMA.
// One wave computes a 16-row stripe across all 128 output columns.
// A layout (16x4 f32, ISA 7.12.2): lanes 0-15 hold rows M=0..15; VGPR0/1 =
//   K=0/1 for lanes 0-15, K=2/3 for lanes 16-31.
// B layout (4x16 f32, mirrored): lane n holds column N=n (n&15); VGPR0/1 =
//   K=0/1 (lanes 0-15), K=2/3 (lanes 16-31).
// C/D (16x16 f32): VGPR r = row (r + 8*half), col = lane&15.
// ---------------------------------------------------------------------------
__global__ __launch_bounds__(256) void gat_gemm(const float* __restrict__ x,
                                                const float* __restrict__ W,
                                                float* __restrict__ y) {
    const int wave  = (blockIdx.x * blockDim.x + threadIdx.x) >> 5;
    const int lane  = threadIdx.x & 31;
    const int l16   = lane & 15;
    const int lhalf = lane >> 4;          // 0: K 0/1, 1: K 2/3
    const int m0    = wave * 16;

    const float* __restrict__ xrow = x + (size_t)(m0 + l16) * FDIM;

    v8f c[8] = {};                        // 8 column tiles of 16x16 f32

    for (int k = 0; k < FDIM; k += 4) {
        v2f a;
        a.x = xrow[k + lhalf * 2 + 0];
        a.y = xrow[k + lhalf * 2 + 1];
        const float* __restrict__ wr = W + (size_t)(k + lhalf * 2) * FDIM + l16;
#pragma unroll
        for (int t = 0; t < 8; ++t) {
            v2f b;
            b.x = wr[t * 16];
            b.y = wr[t * 16 + FDIM];
            // (neg_a, A, neg_b, B, c_mod, C, reuse_a, reuse_b)
            c[t] = __builtin_amdgcn_wmma_f32_16x16x4_f32(
                false, a, false, b, (short)0, c[t], false, false);
        }
    }

    // Store: c[t][r] -> y[m0 + r + 8*lhalf][t*16 + l16]
#pragma unroll
    for (int t = 0; t < 8; ++t) {
        float* __restrict__ yb = y + (size_t)(m0 + lhalf * 8) * FDIM + t * 16 + l16;
#pragma unroll
        for (int r = 0; r < 8; ++r) {
            yb[(size_t)r * FDIM] = c[t][r];
        }
    }
}

// ---------------------------------------------------------------------------
// Kernel 2: s1[n] = y[n,:] . a[0:128],  s2[n] = y[n,:] . a[128:256]
// One wave per row; float4 loads, wave32 xor-shuffle reduction.
// ---------------------------------------------------------------------------
__global__ __launch_bounds__(256) void gat_rowdots(const float* __restrict__ y,
                                                   const float* __restrict__ a,
                                                   float* __restrict__ s1,
                                                   float* __restrict__ s2) {
    const int row  = (blockIdx.x * blockDim.x + threadIdx.x) >> 5;
    const int lane = threadIdx.x & 31;

    const float4 v  = ((const float4*)(y + (size_t)row * FDIM))[lane];
    const float4 w1 = ((const float4*)a)[lane];
    const float4 w2 = ((const float4*)(a + FDIM))[lane];

    float d1 = v.x * w1.x + v.y * w1.y + v.z * w1.z + v.w * w1.w;
    float d2 = v.x * w2.x + v.y * w2.y + v.z * w2.z + v.w * w2.w;
#pragma unroll
    for (int m = 16; m >= 1; m >>= 1) {
        d1 += __shfl_xor(d1, m, 32);
        d2 += __shfl_xor(d2, m, 32);
    }
    if (lane == 0) {
        s1[row] = d1;
        s2[row] = d2;
    }
}

// ---------------------------------------------------------------------------
// Kernel 3: per node: leaky-relu logits from s1/s2 gathers, softmax over the
// 16 neighbors (wave32 shuffles), weighted aggregation of y rows, ELU.
// One wave per node; each lane owns 4 of 128 output features.
// ---------------------------------------------------------------------------
__global__ __launch_bounds__(256) void gat_attn(const long long* __restrict__ adj,
                                                const float* __restrict__ y,
                                                const float* __restrict__ s1,
                                                const float* __restrict__ s2,
                                                float* __restrict__ out) {
    const int node = (blockIdx.x * blockDim.x + threadIdx.x) >> 5;  // 0..79999
    const int lane = threadIdx.x & 31;
    const int b    = node / L_NODES;
    const int roff = b * L_NODES;     // row offset into y/s for this batch

    int idx = 0;
    if (lane < DEG) idx = (int)adj[(size_t)node * DEG + lane];
    const int idx0 = __shfl(idx, 0, 32);

    float e = -1e30f;
    if (lane < DEG) {
        const float v = s1[roff + idx] + s2[roff + idx0];
        e = (v > 0.f) ? v : LRELU_ALPHA * v;   // leaky_relu(., 0.2)
    }

    // softmax over lanes 0..15 (upper half carries -inf / 0)
    float m = e;
#pragma unroll
    for (int s = 16; s >= 1; s >>= 1) m = fmaxf(m, __shfl_xor(m, s, 32));
    float p = (lane < DEG) ? __expf(e - m) : 0.f;
    float sum = p;
#pragma unroll
    for (int s = 16; s >= 1; s >>= 1) sum += __shfl_xor(sum, s, 32);
    const float attn = p / sum;

    // aggregate: out[node, lane*4 .. lane*4+3] = sum_d attn_d * y[idx_d]
    float4 acc = {0.f, 0.f, 0.f, 0.f};
    const float4* __restrict__ y4 = (const float4*)y;
#pragma unroll
    for (int d = 0; d < DEG; ++d) {
        const float ad = __shfl(attn, d, 32);
        const int   id = __shfl(idx, d, 32);
        const float4 v = y4[(size_t)(roff + id) * (FDIM / 4) + lane];
        acc.x += ad * v.x;
        acc.y += ad * v.y;
        acc.z += ad * v.z;
        acc.w += ad * v.w;
    }

    // ELU (alpha = 1)
    acc.x = (acc.x > 0.f) ? acc.x : __expf(acc.x) - 1.f;
    acc.y = (acc.y > 0.f) ? acc.y : __expf(acc.y) - 1.f;
    acc.z = (acc.z > 0.f) ? acc.z : __expf(acc.z) - 1.f;
    acc.w = (acc.w > 0.f) ? acc.w : __expf(acc.w) - 1.f;

    ((float4*)out)[(size_t)node * (FDIM / 4) + lane] = acc;
}

// ---------------------------------------------------------------------------
extern "C" void kernel_launch(void* const* d_in, const int* in_sizes, int n_in,
                              void* d_out, int out_size, void* d_ws, size_t ws_size,
                              hipStream_t stream) {
    const float*     x   = (const float*)d_in[0];      // (4, 20000, 128) f32
    const long long* adj = (const long long*)d_in[1];  // (4, 20000, 16) i64
    const float*     W   = (const float*)d_in[2];      // (128, 128) f32
    const float*     a   = (const float*)d_in[3];      // (256, 1) f32
    float*           out = (float*)d_out;              // (4, 20000, 128) f32

    float* y  = (float*)d_ws;                          // 80000*128 f32 = 40.96 MB
    float* s1 = y + (size_t)NNODES * FDIM;             // 80000 f32
    float* s2 = s1 + NNODES;                           // 80000 f32

    // 1) y = x @ W   (5000 waves of 16 rows, 8 waves per block)
    gat_gemm<<<NNODES / 16 / 8, 256, 0, stream>>>(x, W, y);
    // 2) s1, s2 row dots (one wave per row)
    gat_rowdots<<<NNODES / 8, 256, 0, stream>>>(y, a, s1, s2);
    // 3) attention + aggregation + ELU (one wave per node)
    gat_attn<<<NNODES / 8, 256, 0, stream>>>(adj, y, s1, s2, out);
}